// FPSPooling_40286793236956
// MI455X (gfx1250) — compile-verified
//
#include <hip/hip_runtime.h>
#include <hip/hip_bf16.h>

typedef __attribute__((ext_vector_type(2))) float v2f;
typedef __attribute__((ext_vector_type(8))) float v8f;

#define FPS_N   16384
#define FPS_E   262144
#define NPOOL   8192
#define NKEYS   (2 * FPS_E)          /* 524288 = 2^19 */
#define SENT    (NPOOL * NPOOL)      /* 67108864 */
#define EDGE_OUT (2 * FPS_E + 1)     /* 524289 */
#define SCAN_B  512

// ---------------------------------------------------------------------------
// Kernel 1: Farthest Point Sampling. Single 1024-thread workgroup; coords +
// min-dist array live entirely in LDS (256KB of the WGP's 320KB). Distance
// updates are done with V_WMMA_F32_16X16X4_F32: A = broadcast (-2cx,-2cy,-2cz,1),
// B columns = (x,y,z,|x|^2) for 16 points -> C row 0 = |x|^2 - 2 c.x per lane.
// ---------------------------------------------------------------------------
__global__ __launch_bounds__(1024)
void fps_kernel(const float* __restrict__ x, int* __restrict__ node_out) {
    extern __shared__ char smem[];
    float* sx = (float*)smem;
    float* sy = sx + FPS_N;
    float* sz = sy + FPS_N;
    float* sd = sz + FPS_N;
    unsigned long long* skey = (unsigned long long*)(sd + FPS_N);

    const int tid  = threadIdx.x;
    const int lane = tid & 31;
    const int wave = tid >> 5;

    for (int i = tid; i < FPS_N; i += 1024) {
        sx[i] = x[3 * i + 0];
        sy[i] = x[3 * i + 1];
        sz[i] = x[3 * i + 2];
        sd[i] = 3.402823466e38f;
    }
    if (tid == 0) node_out[0] = 0;
    __syncthreads();

    float cx = sx[0], cy = sy[0], cz = sz[0];
    const bool lo  = (lane < 16);
    const int  l15 = lane & 15;

    for (int s = 1; s < NPOOL; ++s) {
        // ---- distance update against current center (cx,cy,cz) via WMMA ----
        {
            const float csq = cx * cx + cy * cy + cz * cz;
            v2f a;
            a.x = lo ? (-2.0f * cx) : (-2.0f * cz);
            a.y = lo ? (-2.0f * cy) : 1.0f;
            v8f zc = {};
            #pragma unroll
            for (int t = 0; t < 16; ++t) {
                const int base = (wave << 9) + (t << 5);  // wave*512 + t*32
                const int p1 = base + l15;
                const int p2 = p1 + 16;
                const float x1 = sx[p1], y1 = sy[p1], z1 = sz[p1];
                const float x2 = sx[p2], y2 = sy[p2], z2 = sz[p2];
                v2f b1, b2;
                b1.x = lo ? x1 : z1;
                b1.y = lo ? y1 : (x1 * x1 + y1 * y1 + z1 * z1);
                b2.x = lo ? x2 : z2;
                b2.y = lo ? y2 : (x2 * x2 + y2 * y2 + z2 * z2);
                v8f d1 = __builtin_amdgcn_wmma_f32_16x16x4_f32(
                    false, a, false, b1, (short)0, zc, false, false);
                v8f d2 = __builtin_amdgcn_wmma_f32_16x16x4_f32(
                    false, a, false, b2, (short)0, zc, false, false);
                const float dot = lo ? d1[0] : d2[0];
                const float dd  = fmaxf(dot + csq, 0.0f);  // keep >=0 so bit
                const int   p   = base + lane;             // pattern is ordered
                sd[p] = fminf(sd[p], dd);
            }
        }
        if (tid == 0) *skey = 0ULL;
        __syncthreads();

        // ---- argmax over sd: pack (dist_bits<<32) | ~index, max-reduce ----
        unsigned long long key = 0ULL;
        #pragma unroll
        for (int k = 0; k < 16; ++k) {
            const int p = tid + (k << 10);
            const unsigned long long kk =
                ((unsigned long long)__float_as_uint(sd[p]) << 32) |
                (unsigned)(~p);
            key = (key > kk) ? key : kk;
        }
        #pragma unroll
        for (int off = 16; off > 0; off >>= 1) {
            const unsigned hi = __shfl_xor((unsigned)(key >> 32), off, 32);
            const unsigned lw = __shfl_xor((unsigned)(key & 0xffffffffu), off, 32);
            const unsigned long long o = ((unsigned long long)hi << 32) | lw;
            key = (key > o) ? key : o;
        }
        if (lane == 0) atomicMax(skey, key);  // ds_max_u64 on LDS
        __syncthreads();

        const unsigned long long win = *skey;
        const int idx = (int)(~(unsigned)(win & 0xffffffffu));
        if (tid == 0) node_out[s] = idx;
        cx = sx[idx]; cy = sy[idx]; cz = sz[idx];
        __syncthreads();  // everyone has read skey before next reset
    }
}

// ---------------------------------------------------------------------------
// pool_edges pipeline
// ---------------------------------------------------------------------------
__global__ void fill_i32(int* __restrict__ p, int n, int v) {
    const int i = blockIdx.x * blockDim.x + threadIdx.x;
    if (i < n) p[i] = v;
}

__global__ void set_ids(const int* __restrict__ node, int* __restrict__ new_id) {
    const int i = blockIdx.x * blockDim.x + threadIdx.x;
    if (i < NPOOL) new_id[node[i]] = i;
}

__global__ void build_keys(const int* __restrict__ ei,
                           const int* __restrict__ new_id,
                           int* __restrict__ keys) {
    const int i = blockIdx.x * blockDim.x + threadIdx.x;
    if (i >= FPS_E) return;
    __builtin_prefetch(&ei[FPS_E + i], 0, 0);
    const int s = ei[i];
    const int d = ei[FPS_E + i];
    const int r = new_id[s];
    const int c = new_id[d];
    const bool valid = (r >= 0) && (c >= 0);
    keys[i]         = valid ? (r * NPOOL + c) : SENT;
    keys[FPS_E + i] = valid ? (c * NPOOL + r) : SENT;
}

__global__ void bitonic_step(int* __restrict__ a, int j, int k) {
    const int i   = blockIdx.x * blockDim.x + threadIdx.x;
    const int ixj = i ^ j;
    if (ixj > i) {
        const int ai = a[i], aj = a[ixj];
        const bool up = ((i & k) == 0);
        if (up ? (ai > aj) : (ai < aj)) { a[i] = aj; a[ixj] = ai; }
    }
}

__global__ __launch_bounds__(SCAN_B)
void scan_blocks(const int* __restrict__ keys, int* __restrict__ pos,
                 int* __restrict__ bsum) {
    __shared__ int tmp[SCAN_B];
    const int tid = threadIdx.x;
    const int j = blockIdx.x * SCAN_B + tid;
    const int f = (j == 0) ? 1 : ((keys[j] != keys[j - 1]) ? 1 : 0);
    tmp[tid] = f;
    __syncthreads();
    for (int off = 1; off < SCAN_B; off <<= 1) {
        const int v = (tid >= off) ? tmp[tid - off] : 0;
        __syncthreads();
        tmp[tid] += v;
        __syncthreads();
    }
    pos[j] = tmp[tid] - f;  // exclusive within block
    if (tid == SCAN_B - 1) bsum[blockIdx.x] = tmp[tid];
}

__global__ __launch_bounds__(1024)
void scan_bsum(int* __restrict__ bsum) {  // NKEYS/SCAN_B == 1024 entries
    __shared__ int tmp[1024];
    const int tid = threadIdx.x;
    const int v0 = bsum[tid];
    tmp[tid] = v0;
    __syncthreads();
    for (int off = 1; off < 1024; off <<= 1) {
        const int v = (tid >= off) ? tmp[tid - off] : 0;
        __syncthreads();
        tmp[tid] += v;
        __syncthreads();
    }
    bsum[tid] = tmp[tid] - v0;  // exclusive
}

__global__ __launch_bounds__(SCAN_B)
void scatter_unique(const int* __restrict__ keys, const int* __restrict__ pos,
                    const int* __restrict__ bsum, int* __restrict__ out) {
    const int j = blockIdx.x * SCAN_B + threadIdx.x;
    const int key = keys[j];
    const int f = (j == 0) ? 1 : (key != keys[j - 1]);
    if (f && key != SENT) {
        const int p = pos[j] + bsum[j / SCAN_B];
        out[p]            = key >> 13;           // row = key / 8192
        out[EDGE_OUT + p] = key & (NPOOL - 1);   // col = key % 8192
    }
}

// ---------------------------------------------------------------------------
extern "C" void kernel_launch(void* const* d_in, const int* in_sizes, int n_in,
                              void* d_out, int out_size, void* d_ws,
                              size_t ws_size, hipStream_t stream) {
    (void)in_sizes; (void)n_in; (void)out_size; (void)ws_size;
    const float* x  = (const float*)d_in[0];
    const int*   ei = (const int*)d_in[1];

    int* out      = (int*)d_out;
    int* edge_out = out;                  // [2, EDGE_OUT] row-major
    int* node_out = out + 2 * EDGE_OUT;   // [NPOOL]

    int* ws     = (int*)d_ws;
    int* keys   = ws;                     // NKEYS ints
    int* pos    = ws + NKEYS;             // NKEYS ints
    int* bsum   = ws + 2 * NKEYS;         // 1024 ints
    int* new_id = ws + 2 * NKEYS + 1024;  // FPS_N ints

    // 1. FPS: one persistent workgroup, 256KB+ of dynamic LDS
    fps_kernel<<<1, 1024, 4 * FPS_N * sizeof(float) + 16, stream>>>(x, node_out);

    // 2. prefill pooled edge output with -1 (padding / sentinel decode)
    fill_i32<<<(2 * EDGE_OUT + 255) / 256, 256, 0, stream>>>(edge_out,
                                                             2 * EDGE_OUT, -1);

    // 3. relabel table
    fill_i32<<<(FPS_N + 255) / 256, 256, 0, stream>>>(new_id, FPS_N, -1);
    set_ids<<<(NPOOL + 255) / 256, 256, 0, stream>>>(node_out, new_id);

    // 4. forward + reversed keys (sentinel for invalid edges)
    build_keys<<<(FPS_E + 255) / 256, 256, 0, stream>>>(ei, new_id, keys);

    // 5. bitonic sort of 2^19 keys (190 deterministic passes, L2 resident)
    for (int k = 2; k <= NKEYS; k <<= 1)
        for (int j = k >> 1; j > 0; j >>= 1)
            bitonic_step<<<NKEYS / 256, 256, 0, stream>>>(keys, j, k);

    // 6. dedupe + compact: flag-scan then scatter decoded (r,c)
    scan_blocks<<<NKEYS / SCAN_B, SCAN_B, 0, stream>>>(keys, pos, bsum);
    scan_bsum<<<1, 1024, 0, stream>>>(bsum);
    scatter_unique<<<NKEYS / SCAN_B, SCAN_B, 0, stream>>>(keys, pos, bsum,
                                                          edge_out);
}